// Recommender_6949257085118
// MI455X (gfx1250) — compile-verified
//
#include <hip/hip_runtime.h>

typedef float v2f __attribute__((ext_vector_type(2)));
typedef float v8f __attribute__((ext_vector_type(8)));

#define HD 128
constexpr int N_USER = 150000;
constexpr int N_BOOK = 75000;
constexpr int N_EDGE = 1500000;
constexpr float LN_EPS = 1e-5f;

// ---------------------------------------------------------------- utilities
__global__ void zero_kernel(float* __restrict__ p, size_t n) {
    size_t i = (size_t)blockIdx.x * blockDim.x + threadIdx.x;
    size_t stride = (size_t)gridDim.x * blockDim.x;
    for (; i < n; i += stride) p[i] = 0.0f;
}

__global__ void count_kernel(const int* __restrict__ esrc, const int* __restrict__ edst,
                             float* __restrict__ cnt_u, float* __restrict__ cnt_b, int E) {
    int i = blockIdx.x * blockDim.x + threadIdx.x;
    if (i < E) {
        atomicAdd(&cnt_u[esrc[i]], 1.0f);
        atomicAdd(&cnt_b[edst[i]], 1.0f);
    }
}

// One wave32 per edge: lanes cover the full 128-float row (float4 each).
__global__ void scatter_kernel(const float* __restrict__ feat, const int* __restrict__ sidx,
                               const int* __restrict__ didx, float* __restrict__ sums, int E) {
    int gid = blockIdx.x * blockDim.x + threadIdx.x;
    int e = gid >> 5;
    if (e >= E) return;
    int lane = gid & 31;
    int s = sidx[e], d = didx[e];
    const float4 v = *(const float4*)(feat + (size_t)s * HD + lane * 4);
    float* dp = sums + (size_t)d * HD + lane * 4;
    atomicAdd(dp + 0, v.x);
    atomicAdd(dp + 1, v.y);
    atomicAdd(dp + 2, v.z);
    atomicAdd(dp + 3, v.w);
}

// ---------------------------------------------------------------- WMMA GEMMs
// fp32 WMMA 16x16x4. Per-lane operand layout (wave32):
//   A 16x4 : row M = lane%16 ; v2f = {A[m, k0+2*(lane/16)], A[m, k0+2*(lane/16)+1]}
//   B 4x16 : col N = lane%16 ; v2f = {B[k0+2*(lane/16), n], B[k0+2*(lane/16)+1, n]}
//   C 16x16: col N = lane%16 ; component v holds row M = v + 8*(lane/16)
//
// Weights live in LDS *transposed* (wt[n][k], row stride K+4) so the B fragment
// is ONE aligned ds_load_b64; stride % 64 == 4 makes the 32-lane b64 read hit
// all 64 banks exactly once (conflict-free).

// cooperative global->LDS transposed weight load (256 threads)
__device__ __forceinline__ void load_weight_t(const float* __restrict__ W, float* __restrict__ wt,
                                              int K, int stride) {
    const int nf4 = K * (HD / 4);
    for (int idx = threadIdx.x; idx < nf4; idx += 256) {
        int k = idx >> 5;          // 32 float4 per 128-wide row
        int n = (idx & 31) << 2;
        const float4 w4 = *(const float4*)(W + (size_t)k * HD + n);
        wt[(n + 0) * stride + k] = w4.x;
        wt[(n + 1) * stride + k] = w4.y;
        wt[(n + 2) * stride + k] = w4.z;
        wt[(n + 3) * stride + k] = w4.w;
    }
}

// out[M,128] = A[M,K] @ W[K,128] + bias   (projection; K = 64 or 128)
__global__ void gemm_bias_kernel(const float* __restrict__ A, const float* __restrict__ W,
                                 const float* __restrict__ bias, float* __restrict__ out,
                                 int M, int K) {
    __shared__ float wt[HD * (HD + 4)];
    const int stride = K + 4;
    load_weight_t(W, wt, K, stride);
    __syncthreads();

    const int wave = threadIdx.x >> 5;          // 8 waves / block -> 128 rows
    const int lane = threadIdx.x & 31;
    const int col  = lane & 15;
    const int half = lane >> 4;
    const int m0   = blockIdx.x * 128 + wave * 16;
    const int aRow = (m0 + col < M) ? (m0 + col) : (M - 1);

    v8f acc[8];
#pragma unroll
    for (int nt = 0; nt < 8; ++nt) {
        float bv = bias[nt * 16 + col];
        acc[nt] = (v8f){bv, bv, bv, bv, bv, bv, bv, bv};
    }

    const float* arow = A + (size_t)aRow * K;
    for (int k0 = 0; k0 < K; k0 += 4) {
        const int kb = k0 + 2 * half;
        v2f a = *(const v2f*)(arow + kb);
        v2f bf[8];
#pragma unroll
        for (int nt = 0; nt < 8; ++nt)
            bf[nt] = *(const v2f*)&wt[(nt * 16 + col) * stride + kb];
#pragma unroll
        for (int nt = 0; nt < 8; ++nt)
            acc[nt] = __builtin_amdgcn_wmma_f32_16x16x4_f32(
                false, a, false, bf[nt], (short)0, acc[nt], false, false);
    }

#pragma unroll
    for (int nt = 0; nt < 8; ++nt)
#pragma unroll
        for (int v = 0; v < 8; ++v) {
            int dr = m0 + v + 8 * half;
            if (dr < M) out[(size_t)dr * HD + nt * 16 + col] = acc[nt][v];
        }
}

// out = relu( (sums/max(cnt,1)) @ Wl + bl + X @ Wr )
// Single 67.6KB LDS buffer reused for Wl then Wr (phase barriers).
__global__ void sage_combine_kernel(const float* __restrict__ sums, const float* __restrict__ cnt,
                                    const float* __restrict__ X, const float* __restrict__ Wl,
                                    const float* __restrict__ bl, const float* __restrict__ Wr,
                                    float* __restrict__ out, int M) {
    __shared__ float wt[HD * (HD + 4)];
    const int stride = HD + 4;

    load_weight_t(Wl, wt, HD, stride);
    __syncthreads();

    const int wave = threadIdx.x >> 5;
    const int lane = threadIdx.x & 31;
    const int col  = lane & 15;
    const int half = lane >> 4;
    const int m0   = blockIdx.x * 128 + wave * 16;
    const int aRow = (m0 + col < M) ? (m0 + col) : (M - 1);
    const float invc = 1.0f / fmaxf(cnt[aRow], 1.0f);

    v8f acc[8];
#pragma unroll
    for (int nt = 0; nt < 8; ++nt) {
        float bv = bl[nt * 16 + col];
        acc[nt] = (v8f){bv, bv, bv, bv, bv, bv, bv, bv};
    }

    // phase 1: mean @ Wl
    const float* srow = sums + (size_t)aRow * HD;
    for (int k0 = 0; k0 < HD; k0 += 4) {
        const int kb = k0 + 2 * half;
        v2f a = *(const v2f*)(srow + kb);
        a.x *= invc;
        a.y *= invc;
        v2f bf[8];
#pragma unroll
        for (int nt = 0; nt < 8; ++nt)
            bf[nt] = *(const v2f*)&wt[(nt * 16 + col) * stride + kb];
#pragma unroll
        for (int nt = 0; nt < 8; ++nt)
            acc[nt] = __builtin_amdgcn_wmma_f32_16x16x4_f32(
                false, a, false, bf[nt], (short)0, acc[nt], false, false);
    }

    __syncthreads();               // everyone done reading Wl
    load_weight_t(Wr, wt, HD, stride);
    __syncthreads();

    // phase 2: X @ Wr
    const float* xrow = X + (size_t)aRow * HD;
    for (int k0 = 0; k0 < HD; k0 += 4) {
        const int kb = k0 + 2 * half;
        v2f a = *(const v2f*)(xrow + kb);
        v2f bf[8];
#pragma unroll
        for (int nt = 0; nt < 8; ++nt)
            bf[nt] = *(const v2f*)&wt[(nt * 16 + col) * stride + kb];
#pragma unroll
        for (int nt = 0; nt < 8; ++nt)
            acc[nt] = __builtin_amdgcn_wmma_f32_16x16x4_f32(
                false, a, false, bf[nt], (short)0, acc[nt], false, false);
    }

#pragma unroll
    for (int nt = 0; nt < 8; ++nt)
#pragma unroll
        for (int v = 0; v < 8; ++v) {
            int dr = m0 + v + 8 * half;
            if (dr < M) out[(size_t)dr * HD + nt * 16 + col] = fmaxf(acc[nt][v], 0.0f);
        }
}

// ---------------------------------------------------------------- layernorm
// one wave per 128-float row; wave32 shfl-xor reductions
__global__ void layernorm_kernel(const float* __restrict__ X, const float* __restrict__ gamma,
                                 const float* __restrict__ beta, float* __restrict__ out, int M) {
    const int wave = threadIdx.x >> 5;
    const int lane = threadIdx.x & 31;
    const int row  = blockIdx.x * 8 + wave;
    if (row >= M) return;

    const float4 v = *(const float4*)(X + (size_t)row * HD + lane * 4);
    float s = v.x + v.y + v.z + v.w;
#pragma unroll
    for (int off = 16; off >= 1; off >>= 1) s += __shfl_xor(s, off, 32);
    const float mu = s * (1.0f / HD);

    const float dx = v.x - mu, dy = v.y - mu, dz = v.z - mu, dw = v.w - mu;
    float q = dx * dx + dy * dy + dz * dz + dw * dw;
#pragma unroll
    for (int off = 16; off >= 1; off >>= 1) q += __shfl_xor(q, off, 32);
    const float rstd = rsqrtf(q * (1.0f / HD) + LN_EPS);

    const float4 g = *(const float4*)(gamma + lane * 4);
    const float4 bt = *(const float4*)(beta + lane * 4);
    float4 o;
    o.x = dx * rstd * g.x + bt.x;
    o.y = dy * rstd * g.y + bt.y;
    o.z = dz * rstd * g.z + bt.z;
    o.w = dw * rstd * g.w + bt.w;
    *(float4*)(out + (size_t)row * HD + lane * 4) = o;
}

// ---------------------------------------------------------------- launcher
extern "C" void kernel_launch(void* const* d_in, const int* in_sizes, int n_in,
                              void* d_out, int out_size, void* d_ws, size_t ws_size,
                              hipStream_t stream) {
    const float* user_x      = (const float*)d_in[0];
    const float* book_x      = (const float*)d_in[1];
    const int*   edge_src    = (const int*)d_in[2];
    const int*   edge_dst    = (const int*)d_in[3];
    const float* user_proj_w = (const float*)d_in[4];
    const float* user_proj_b = (const float*)d_in[5];
    const float* book_proj_w = (const float*)d_in[6];
    const float* book_proj_b = (const float*)d_in[7];
    const float* Wl          = (const float*)d_in[8];
    const float* bl          = (const float*)d_in[9];
    const float* Wr          = (const float*)d_in[10];
    const float* ugamma      = (const float*)d_in[11];
    const float* ubeta       = (const float*)d_in[12];
    const float* bgamma      = (const float*)d_in[13];
    const float* bbeta       = (const float*)d_in[14];

    float* ws = (float*)d_ws;
    size_t off = 0;
    auto carve = [&](size_t n) { float* p = ws + off; off += n; return p; };
    float* u_cur = carve((size_t)N_USER * HD);
    float* u_nxt = carve((size_t)N_USER * HD);
    float* b_cur = carve((size_t)N_BOOK * HD);
    float* b_nxt = carve((size_t)N_BOOK * HD);
    float* sum_u = carve((size_t)N_USER * HD);
    float* sum_b = carve((size_t)N_BOOK * HD);
    float* cnt_u = carve((size_t)N_USER);
    float* cnt_b = carve((size_t)N_BOOK);

    // projections (fp32 WMMA GEMM + bias)
    gemm_bias_kernel<<<(N_USER + 127) / 128, 256, 0, stream>>>(
        user_x, user_proj_w, user_proj_b, u_cur, N_USER, 64);
    gemm_bias_kernel<<<(N_BOOK + 127) / 128, 256, 0, stream>>>(
        book_x, book_proj_w, book_proj_b, b_cur, N_BOOK, 128);

    // degree counts (fixed across layers)
    zero_kernel<<<256, 256, 0, stream>>>(cnt_u, (size_t)N_USER);
    zero_kernel<<<256, 256, 0, stream>>>(cnt_b, (size_t)N_BOOK);
    count_kernel<<<(N_EDGE + 255) / 256, 256, 0, stream>>>(
        edge_src, edge_dst, cnt_u, cnt_b, N_EDGE);

    const int scat_blocks = (N_EDGE * 32 + 255) / 256;
    for (int l = 0; l < 3; ++l) {
        const float* wl  = Wl + (size_t)l * HD * HD;
        const float* blv = bl + (size_t)l * HD;
        const float* wr  = Wr + (size_t)l * HD * HD;

        zero_kernel<<<4096, 256, 0, stream>>>(sum_b, (size_t)N_BOOK * HD);
        zero_kernel<<<4096, 256, 0, stream>>>(sum_u, (size_t)N_USER * HD);

        // messages use OLD features in both directions
        scatter_kernel<<<scat_blocks, 256, 0, stream>>>(u_cur, edge_src, edge_dst, sum_b, N_EDGE);
        scatter_kernel<<<scat_blocks, 256, 0, stream>>>(b_cur, edge_dst, edge_src, sum_u, N_EDGE);

        sage_combine_kernel<<<(N_BOOK + 127) / 128, 256, 0, stream>>>(
            sum_b, cnt_b, b_cur, wl, blv, wr, b_nxt, N_BOOK);
        sage_combine_kernel<<<(N_USER + 127) / 128, 256, 0, stream>>>(
            sum_u, cnt_u, u_cur, wl, blv, wr, u_nxt, N_USER);

        float* t;
        t = u_cur; u_cur = u_nxt; u_nxt = t;
        t = b_cur; b_cur = b_nxt; b_nxt = t;
    }

    float* out_u = (float*)d_out;
    float* out_b = out_u + (size_t)N_USER * HD;
    layernorm_kernel<<<(N_USER + 7) / 8, 256, 0, stream>>>(u_cur, ugamma, ubeta, out_u, N_USER);
    layernorm_kernel<<<(N_BOOK + 7) / 8, 256, 0, stream>>>(b_cur, bgamma, bbeta, out_b, N_BOOK);
}